// GCN_35124242547073
// MI455X (gfx1250) — compile-verified
//
#include <hip/hip_runtime.h>
#include <hip/hip_bf16.h>

// ---------------------------------------------------------------------------
// GCN (3-layer) for MI455X / gfx1250, wave32, WMMA fp32 path.
// ---------------------------------------------------------------------------

#define N_NODES  50000
#define N_EDGES  800000
#define IN_FEAT  128
#define HID2     256   // 2*HIDDEN (W1 output)
#define HIDDEN   128
#define N_CLS    32

typedef __attribute__((ext_vector_type(2))) float v2f;
typedef __attribute__((ext_vector_type(8))) float v8f;

// Relaxed agent-scope fp32 atomic add (no ordering fences; accumulation is
// order-independent up to fp rounding). Should lower to global_atomic_add_f32.
__device__ __forceinline__ void atomic_add_rlx(float* p, float v) {
    __hip_atomic_fetch_add(p, v, __ATOMIC_RELAXED, __HIP_MEMORY_SCOPE_AGENT);
}

// ---------------------------------------------------------------------------
// Degree / norm precompute
// ---------------------------------------------------------------------------
__global__ __launch_bounds__(256) void deg_init_kernel(float* __restrict__ deg, int n) {
    int i = blockIdx.x * blockDim.x + threadIdx.x;
    if (i < n) deg[i] = 1.0f;   // self-loop contribution
}

__global__ __launch_bounds__(256) void deg_accum_kernel(const int* __restrict__ dst,
                                                        float* __restrict__ deg, int e) {
    int i = blockIdx.x * blockDim.x + threadIdx.x;
    if (i < e) atomic_add_rlx(&deg[dst[i]], 1.0f);  // exact: integer-valued < 2^24
}

__global__ __launch_bounds__(256) void dinv_kernel(const float* __restrict__ deg,
                                                   float* __restrict__ dinv, int n) {
    int i = blockIdx.x * blockDim.x + threadIdx.x;
    if (i < n) dinv[i] = rsqrtf(deg[i]);   // deg >= 1 always (self-loop)
}

__global__ __launch_bounds__(256) void norm_kernel(const int* __restrict__ src,
                                                   const int* __restrict__ dst,
                                                   const float* __restrict__ dinv,
                                                   float* __restrict__ norm, int e) {
    int i = blockIdx.x * blockDim.x + threadIdx.x;
    if (i < e) norm[i] = dinv[src[i]] * dinv[dst[i]];
}

// ---------------------------------------------------------------------------
// WMMA fp32 GEMM: C[M x Nc] = A[M x K] @ B[K x Nc], all row-major fp32.
// Each wave computes NT consecutive 16x16 tiles along N (A fragment reuse).
// V_WMMA_F32_16X16X4_F32, K stepped by 4.
// Fragment layouts per CDNA5 ISA 7.12.2 (fp32, wave32):
//   A 16x4 : lanes 0-15 row M=lane, VGPR0=K0,VGPR1=K1; lanes 16-31: K2,K3.
//   B 4x16 : lanes 0-15 col N=lane, VGPR0=K0 row,VGPR1=K1; lanes 16-31: K2,K3.
//   C 16x16: VGPR r -> row r (lanes 0-15) / row 8+r (lanes 16-31), col = lane&15.
// ---------------------------------------------------------------------------
template <int NT>
__global__ __launch_bounds__(128) void wmma_gemm_f32_kernel(
    const float* __restrict__ A, const float* __restrict__ B, float* __restrict__ C,
    int M, int K, int Nc) {
    const int lane   = threadIdx.x & 31;
    const int wave   = threadIdx.x >> 5;
    const int mtile  = blockIdx.x;
    const int ntile0 = (blockIdx.y * 4 + wave) * NT;

    if (ntile0 * 16 >= Nc) return;         // wave-uniform guard (EXEC stays all-1)
    const int row0 = mtile * 16;
    if (row0 >= M) return;

    const int lhalf = lane >> 4;           // 0: K pair {0,1}, 1: K pair {2,3}
    const int lmod  = lane & 15;

    const float* Aptr = A + (long)(row0 + lmod) * K + 2 * lhalf;
    const float* Bptr = B + (long)(2 * lhalf) * Nc + ntile0 * 16 + lmod;

    v8f acc[NT];
#pragma unroll
    for (int t = 0; t < NT; ++t) acc[t] = (v8f){};

    for (int k0 = 0; k0 < K; k0 += 4) {
        const float2 av = *(const float2*)(Aptr + k0);
        v2f a; a.x = av.x; a.y = av.y;
#pragma unroll
        for (int t = 0; t < NT; ++t) {
            v2f b;
            b.x = Bptr[(long)k0 * Nc + t * 16];
            b.y = Bptr[(long)(k0 + 1) * Nc + t * 16];
            acc[t] = __builtin_amdgcn_wmma_f32_16x16x4_f32(
                /*neg_a=*/false, a, /*neg_b=*/false, b,
                /*c_mod=*/(short)0, acc[t], /*reuse_a=*/false, /*reuse_b=*/false);
        }
    }

    float* Cptr = C + (long)(row0 + 8 * lhalf) * Nc + ntile0 * 16 + lmod;
#pragma unroll
    for (int t = 0; t < NT; ++t) {
#pragma unroll
        for (int r = 0; r < 8; ++r) {
            Cptr[(long)r * Nc + t * 16] = acc[t][r];
        }
    }
}

// ---------------------------------------------------------------------------
// Aggregation: agg = D^-1/2 (A+I) D^-1/2 h, split into self-loop init + edges.
// ---------------------------------------------------------------------------
template <int F>
__global__ __launch_bounds__(256) void self_init_kernel(const float* __restrict__ h,
                                                        const float* __restrict__ dinv,
                                                        float* __restrict__ agg, int n) {
    long tid = (long)blockIdx.x * blockDim.x + threadIdx.x;
    if (tid >= (long)n * F) return;
    int i = (int)(tid / F);
    float w = dinv[i];
    agg[tid] = w * w * h[tid];
}

template <int F>
__global__ __launch_bounds__(256) void edge_agg_kernel(const float* __restrict__ h,
                                                       const int* __restrict__ src,
                                                       const int* __restrict__ dst,
                                                       const float* __restrict__ norm,
                                                       float* __restrict__ agg, int e) {
    constexpr int FG = F / 4;
    long tid = (long)blockIdx.x * blockDim.x + threadIdx.x;
    if (tid >= (long)e * FG) return;
    int ed = (int)(tid / FG);
    int f  = (int)(tid % FG) * 4;
    int s = src[ed], d = dst[ed];
    float w = norm[ed];
    const float4 hv = *(const float4*)(h + (long)s * F + f);
    float* o = agg + (long)d * F + f;
    atomic_add_rlx(o + 0, w * hv.x);
    atomic_add_rlx(o + 1, w * hv.y);
    atomic_add_rlx(o + 2, w * hv.z);
    atomic_add_rlx(o + 3, w * hv.w);
}

template <int F, bool RELU>
__global__ __launch_bounds__(256) void bias_act_kernel(float* __restrict__ a,
                                                       const float* __restrict__ b, int n) {
    long tid = (long)blockIdx.x * blockDim.x + threadIdx.x;
    if (tid >= (long)n * F) return;
    int f = (int)(tid % F);
    float v = a[tid] + b[f];
    if (RELU) v = fmaxf(v, 0.0f);
    a[tid] = v;
}

// ---------------------------------------------------------------------------
// Host-side launch
// ---------------------------------------------------------------------------
static inline int cdiv(long a, long b) { return (int)((a + b - 1) / b); }

extern "C" void kernel_launch(void* const* d_in, const int* in_sizes, int n_in,
                              void* d_out, int out_size, void* d_ws, size_t ws_size,
                              hipStream_t stream) {
    const float* x  = (const float*)d_in[0];
    const int*   ei = (const int*)d_in[1];     // [2, E] int32 (JAX x64 disabled)
    const float* W1 = (const float*)d_in[2];
    const float* b1 = (const float*)d_in[3];
    const float* W2 = (const float*)d_in[4];
    const float* b2 = (const float*)d_in[5];
    const float* W3 = (const float*)d_in[6];
    const float* b3 = (const float*)d_in[7];
    float* out = (float*)d_out;

    const int N = N_NODES, E = N_EDGES;
    const int* src = ei;
    const int* dst = ei + E;

    // Workspace layout (floats)
    float* deg  = (float*)d_ws;          // N
    float* dinv = deg  + N;              // N
    float* norm = dinv + N;              // E
    float* Hbuf = norm + E;              // N * 256 (GEMM output, max width)
    float* Abuf = Hbuf + (long)N * HID2; // N * 256 (aggregation output)

    // ---- normalization precompute ----
    deg_init_kernel<<<cdiv(N, 256), 256, 0, stream>>>(deg, N);
    deg_accum_kernel<<<cdiv(E, 256), 256, 0, stream>>>(dst, deg, E);
    dinv_kernel<<<cdiv(N, 256), 256, 0, stream>>>(deg, dinv, N);
    norm_kernel<<<cdiv(E, 256), 256, 0, stream>>>(src, dst, dinv, norm, E);

    const int mtiles = cdiv(N, 16);      // 3125 (exact)

    // ---- layer 1: X[N,128] @ W1[128,256] -> agg -> +b1, ReLU ----
    {
        dim3 grid(mtiles, cdiv(HID2 / 16, 4 * 4));       // NT=4: 4 waves cover 16 ntiles
        wmma_gemm_f32_kernel<4><<<grid, 128, 0, stream>>>(x, W1, Hbuf, N, IN_FEAT, HID2);
        self_init_kernel<HID2><<<cdiv((long)N * HID2, 256), 256, 0, stream>>>(Hbuf, dinv, Abuf, N);
        edge_agg_kernel<HID2><<<cdiv((long)E * (HID2 / 4), 256), 256, 0, stream>>>(
            Hbuf, src, dst, norm, Abuf, E);
        bias_act_kernel<HID2, true><<<cdiv((long)N * HID2, 256), 256, 0, stream>>>(Abuf, b1, N);
    }

    // ---- layer 2: A[N,256] @ W2[256,128] -> agg -> +b2, ReLU ----
    {
        dim3 grid(mtiles, cdiv(HIDDEN / 16, 4 * 2));     // NT=2: 4 waves cover 8 ntiles
        wmma_gemm_f32_kernel<2><<<grid, 128, 0, stream>>>(Abuf, W2, Hbuf, N, HID2, HIDDEN);
        self_init_kernel<HIDDEN><<<cdiv((long)N * HIDDEN, 256), 256, 0, stream>>>(Hbuf, dinv, Abuf, N);
        edge_agg_kernel<HIDDEN><<<cdiv((long)E * (HIDDEN / 4), 256), 256, 0, stream>>>(
            Hbuf, src, dst, norm, Abuf, E);
        bias_act_kernel<HIDDEN, true><<<cdiv((long)N * HIDDEN, 256), 256, 0, stream>>>(Abuf, b2, N);
    }

    // ---- layer 3: A[N,128] @ W3[128,32] -> agg (into d_out) -> +b3 ----
    {
        dim3 grid(mtiles, cdiv(N_CLS / 16, 4));          // NT=1: 2 live waves, 2 idle
        wmma_gemm_f32_kernel<1><<<grid, 128, 0, stream>>>(Abuf, W3, Hbuf, N, HIDDEN, N_CLS);
        self_init_kernel<N_CLS><<<cdiv((long)N * N_CLS, 256), 256, 0, stream>>>(Hbuf, dinv, out, N);
        edge_agg_kernel<N_CLS><<<cdiv((long)E * (N_CLS / 4), 256), 256, 0, stream>>>(
            Hbuf, src, dst, norm, out, E);
        bias_act_kernel<N_CLS, false><<<cdiv((long)N * N_CLS, 256), 256, 0, stream>>>(out, b3, N);
    }
}